// BIMPM_58626303590866
// MI455X (gfx1250) — compile-verified
//
#include <hip/hip_runtime.h>
#include <hip/hip_bf16.h>
#include <math.h>

// ---------------- problem constants ----------------
constexpr int cB   = 16;
constexpr int cS   = 128;
constexpr int cE   = 300;
constexpr int cCD  = 50;
constexpr int cH   = 100;   // LSTM hidden
constexpr int c4H  = 400;
constexpr int cL   = 20;
constexpr int cCLS = 3;
constexpr int cBS  = cB * cS;       // 2048
constexpr float EPSf = 1e-8f;

typedef float v2f __attribute__((ext_vector_type(2)));
typedef float v8f __attribute__((ext_vector_type(8)));
typedef unsigned int u32x4_t __attribute__((ext_vector_type(4)));
typedef int i32x8_t __attribute__((ext_vector_type(8)));
typedef int i32x4_t __attribute__((ext_vector_type(4)));

// gfx1250 hardware tanh (TRANS op). Inline asm hides the op from the
// scheduler, so pad with v_nop per the TRANS co-execution hazard rule.
__device__ __forceinline__ float tanh_fast(float x) {
  float r;
  asm volatile("v_tanh_f32 %0, %1\n\tv_nop" : "=v"(r) : "v"(x));
  return r;
}
__device__ __forceinline__ float sigf(float x) {
  return fmaf(0.5f, tanh_fast(0.5f * x), 0.5f);
}

// Tensor Data Mover: DMA a 2-D f32 tile (rows x elems_x, row stride
// stride_elems) from global memory into LDS. D# per CDNA5 ISA ch.8.
// clang-23 toolchain: 6-arg builtin (g0 u32x4, g1 i32x8, g2 i32x4,
// g3 i32x4, g4 i32x8, cpol i32).
__device__ __forceinline__ void tdm_load_2d(const void* gaddr, unsigned lds_off,
                                            int elems_x, int rows, int stride_elems) {
  const unsigned long long ga = (unsigned long long)gaddr;
  u32x4_t g0;
  g0[0] = 1u;                                   // count=1, user descriptor
  g0[1] = lds_off;                              // lds_addr (bytes)
  g0[2] = (unsigned)(ga & 0xffffffffu);         // global_addr[31:0]
  g0[3] = (unsigned)((ga >> 32) & 0x01ffffffu)  // global_addr[56:32]
          | (2u << 30);                         // type = 2 (image)
  i32x8_t g1;
  g1[0] = (2 << 16);                            // data_size = 4 bytes
  g1[1] = (elems_x & 0xffff) << 16;             // tensor_dim0[15:0]
  g1[2] = (int)(((unsigned)elems_x >> 16) | (((unsigned)rows & 0xffffu) << 16));
  g1[3] = (int)((((unsigned)rows >> 16) & 0xffffu) | (((unsigned)elems_x & 0xffffu) << 16)); // tile_dim0
  g1[4] = rows & 0xffff;                        // tile_dim1 (tile_dim2 = 0)
  g1[5] = stride_elems;                         // tensor_dim0_stride lo32
  g1[6] = 0;
  g1[7] = 0;
  i32x4_t z4 = {0, 0, 0, 0};
  i32x8_t z8 = {0, 0, 0, 0, 0, 0, 0, 0};
  __builtin_amdgcn_tensor_load_to_lds(g0, g1, z4, z4, z8, 0);
}

// ---------------- workspace layout (floats), reused between word & char levels ----
constexpr long SZ_EMB  = (long)cBS * cE;        // max (padded) input dim
constexpr long SZ_XW   = (long)cBS * c4H;
constexpr long SZ_HS   = (long)cBS * cH;
constexpr long SZ_ATT  = (long)cB * cS * cS;
constexpr long SZ_MV   = (long)cBS * 6 * cL;
constexpr long SZ_WPAD = (long)c4H * 304;       // padded Wih copy (max Epad)

constexpr long O_EMB_P = 0;
constexpr long O_EMB_H = O_EMB_P + SZ_EMB;
constexpr long O_XW    = O_EMB_H + SZ_EMB;          // 4 buffers
constexpr long O_HS    = O_XW + 4 * SZ_XW;          // 4 buffers
constexpr long O_NORM  = O_HS + 4 * SZ_HS;          // 4 x cBS
constexpr long O_ATT   = O_NORM + 4 * (long)cBS;    // 2 buffers
constexpr long O_SUM   = O_ATT + 2 * SZ_ATT;        // 4 x cBS
constexpr long O_FEAT  = O_SUM + 4 * (long)cBS;     // 8 buffers [BS,H]
constexpr long O_MV    = O_FEAT + 8 * SZ_HS;        // 2 buffers
constexpr long O_WX    = O_MV + 2 * SZ_MV;          // [B,4H]
constexpr long O_CX    = O_WX + (long)cB * c4H;     // [B,4H]
constexpr long O_X8    = O_CX + (long)cB * c4H;     // [B,8H]
constexpr long O_FC1   = O_X8 + (long)cB * 2 * c4H; // [B,2H]
constexpr long O_WPADF = O_FC1 + (long)cB * 2 * cH;
constexpr long O_WPADB = O_WPADF + SZ_WPAD;

// =================================================================
// Generic strided/batched fp32 WMMA GEMM:
//   C[b][m,n] = sum_k A[b][m*Ars + k*Acs] * B[b][k*Bks + n*Bns]
// One 16x16 tile per wave32, K blocked by 4 (V_WMMA_F32_16X16X4_F32).
// M/N overflow handled by address clamping (those lanes' results are
// never stored); K tail handled in one guarded iteration after the
// unguarded hot loop. VECA/VECB: k-stride==1 -> 64-bit vector loads.
// =================================================================
template <bool VECA, bool VECB>
__global__ __launch_bounds__(256) void wmma_gemm_k(
    const float* __restrict__ A, long Ars, long Acs,
    const float* __restrict__ Bm, long Bks, long Bns,
    float* __restrict__ C, long ldc,
    int M, int N, int K, long sA, long sB, long sC) {
  const int lane = threadIdx.x & 31;
  const int wave = threadIdx.x >> 5;
  const int tilesN = (N + 15) >> 4;
  const int tilesM = (M + 15) >> 4;
  const int T = tilesM * tilesN;
  const int tile = blockIdx.y * (blockDim.x >> 5) + wave;
  if (tile >= T) return;
  const int tm = tile / tilesN, tn = tile % tilesN;
  const float* Ab = A + (long)blockIdx.x * sA;
  const float* Bb = Bm + (long)blockIdx.x * sB;
  float* Cb = C + (long)blockIdx.x * sC;
  const int m0 = tm * 16, n0 = tn * 16;
  const int l16 = lane & 15, half = lane >> 4;
  // clamped fragment addresses (out-of-range rows/cols are never stored)
  const int mA = (m0 + l16 < M) ? (m0 + l16) : (M - 1);
  const int nB = (n0 + l16 < N) ? (n0 + l16) : (N - 1);
  const float* ap = Ab + (long)mA * Ars + (long)(half * 2) * Acs;
  const float* bp = Bb + (long)nB * Bns + (long)(half * 2) * Bks;
  v8f acc = {};
  int k0 = 0;
#pragma unroll 4
  for (; k0 + 4 <= K; k0 += 4) {
    v2f a, b;
    if (VECA) {
      a = *(const v2f*)(ap + k0);                 // Acs == 1
    } else {
      a.x = ap[(long)k0 * Acs];
      a.y = ap[(long)(k0 + 1) * Acs];
    }
    if (VECB) {
      b = *(const v2f*)(bp + k0);                 // Bks == 1
    } else {
      b.x = bp[(long)k0 * Bks];
      b.y = bp[(long)(k0 + 1) * Bks];
    }
    acc = __builtin_amdgcn_wmma_f32_16x16x4_f32(false, a, false, b, (short)0, acc,
                                                false, false);
  }
  if (k0 < K) {  // K tail (never taken for our padded shapes)
    const int ka = k0 + half * 2;
    v2f a = {0.f, 0.f}, b = {0.f, 0.f};
    if (ka < K)     { a.x = ap[(long)k0 * Acs];       b.x = bp[(long)k0 * Bks]; }
    if (ka + 1 < K) { a.y = ap[(long)(k0 + 1) * Acs]; b.y = bp[(long)(k0 + 1) * Bks]; }
    acc = __builtin_amdgcn_wmma_f32_16x16x4_f32(false, a, false, b, (short)0, acc,
                                                false, false);
  }
  const int col = n0 + l16;
  if (col < N) {
#pragma unroll
    for (int j = 0; j < 8; ++j) {
      const int row = m0 + j + half * 8;
      if (row < M) Cb[(long)row * ldc + col] = acc[j];
    }
  }
}

// =================================================================
// Recurrent LSTM: 4 blocks = (seq p/h) x (dir f/b). All state + Whh in LDS.
// Whh (400x100 f32 = 160 KB) is DMA'd into LDS once by the TDM, then every
// step's z(16x400) = h(16x100) @ Whh^T runs WMMA entirely from LDS.
// Backward runs reversed and stores un-reversed.
// =================================================================
__global__ __launch_bounds__(256) void lstm_rec_k(
    const float* __restrict__ xwPf, const float* __restrict__ xwPb,
    const float* __restrict__ xwHf, const float* __restrict__ xwHb,
    const float* __restrict__ WhhF, const float* __restrict__ bF,
    const float* __restrict__ WhhB, const float* __restrict__ bB,
    float* __restrict__ outPf, float* __restrict__ outPb,
    float* __restrict__ outHf, float* __restrict__ outHb) {
  __shared__ float wbuf[c4H * cH];   // 160 KB: LDS-resident Whh
  __shared__ float hbuf[cB * cH];
  __shared__ float cbuf[cB * cH];
  __shared__ float zbuf[cB * c4H];
  const int dir = blockIdx.x & 1;         // 0 = fwd, 1 = bwd
  const int seq = (blockIdx.x >> 1) & 1;  // 0 = p,   1 = h
  const float* xw  = seq ? (dir ? xwHb : xwHf) : (dir ? xwPb : xwPf);
  float*       out = seq ? (dir ? outHb : outHf) : (dir ? outPb : outPf);
  const float* Whh  = dir ? WhhB : WhhF;
  const float* bias = dir ? bB : bF;
  const int tid = threadIdx.x;
  const int lane = tid & 31, wave = tid >> 5;
  const int l16 = lane & 15, half = lane >> 4;
  const int abase = l16 * cH + half * 2;
  // one wave issues the tensor DMA, waits on TENSORcnt; everyone barriers
  if (tid < 32) {
    tdm_load_2d(Whh, (unsigned)(size_t)wbuf, cH, c4H, cH);
    __builtin_amdgcn_s_wait_tensorcnt(0);
  }
  for (int i = tid; i < cB * cH; i += blockDim.x) { hbuf[i] = 0.f; cbuf[i] = 0.f; }
  __syncthreads();
  for (int t = 0; t < cS; ++t) {
    const int tt = dir ? (cS - 1 - t) : t;
    // z = h @ Whh^T, both operands from LDS
    for (int nt = wave; nt < c4H / 16; nt += 8) {
      const int n0 = nt * 16;
      const int bbase = (n0 + l16) * cH + half * 2;
      v8f acc = {};
#pragma unroll 5
      for (int k0 = 0; k0 < cH; k0 += 4) {   // 100 = 25 * 4, no remainder
        v2f a = *(const v2f*)&hbuf[abase + k0];
        v2f b = *(const v2f*)&wbuf[bbase + k0];
        acc = __builtin_amdgcn_wmma_f32_16x16x4_f32(false, a, false, b, (short)0,
                                                    acc, false, false);
      }
#pragma unroll
      for (int j = 0; j < 8; ++j)
        zbuf[(j + half * 8) * c4H + n0 + l16] = acc[j];
    }
    __syncthreads();
    // gates (hardware v_tanh_f32)
    for (int e = tid; e < cB * cH; e += blockDim.x) {
      const int m = e / cH, j = e % cH;
      const float* xwr = xw + ((long)m * cS + tt) * c4H;
      const float zi = zbuf[m * c4H + j]           + xwr[j]           + bias[j];
      const float zf = zbuf[m * c4H + cH + j]      + xwr[cH + j]      + bias[cH + j];
      const float zg = zbuf[m * c4H + 2 * cH + j]  + xwr[2 * cH + j]  + bias[2 * cH + j];
      const float zo = zbuf[m * c4H + 3 * cH + j]  + xwr[3 * cH + j]  + bias[3 * cH + j];
      const float c = sigf(zf) * cbuf[e] + sigf(zi) * tanh_fast(zg);
      const float h = sigf(zo) * tanh_fast(c);
      cbuf[e] = c;
      hbuf[e] = h;
      out[((long)m * cS + tt) * cH + j] = h;
    }
    __syncthreads();
  }
}

// ---------------- small elementwise kernels ----------------
// gather with zero-padding of the feature dim to Epad (multiple of 4)
__global__ void gather_k(const int* __restrict__ ids, const float* __restrict__ tab,
                         float* __restrict__ out, int E, int Epad) {
  long i = (long)blockIdx.x * blockDim.x + threadIdx.x;
  if (i >= (long)cBS * Epad) return;
  const int e = (int)(i % Epad);
  const int bs = (int)(i / Epad);
  out[i] = (e < E) ? tab[(long)ids[bs] * E + e] : 0.f;
}

// zero-padded copy of Wih (c4H x E) -> (c4H x Epad)
__global__ void padw_k(const float* __restrict__ W, float* __restrict__ out,
                       int E, int Epad) {
  int i = blockIdx.x * blockDim.x + threadIdx.x;
  if (i >= c4H * Epad) return;
  const int k = i % Epad, n = i / Epad;
  out[i] = (k < E) ? W[(long)n * E + k] : 0.f;
}

__global__ void norm_k(const float* __restrict__ v, float* __restrict__ out) {
  int i = blockIdx.x * blockDim.x + threadIdx.x;
  if (i >= cBS) return;
  const float* r = v + (long)i * cH;
  float s = 0.f;
  for (int h = 0; h < cH; ++h) s += r[h] * r[h];
  out[i] = sqrtf(s);
}

__global__ void attdiv_k(float* __restrict__ att, const float* __restrict__ n1,
                         const float* __restrict__ n2) {
  long i = (long)blockIdx.x * blockDim.x + threadIdx.x;
  if (i >= (long)cB * cS * cS) return;
  const int q = (int)(i % cS);
  const long r = i / cS;
  const int p = (int)(r % cS);
  const int b = (int)(r / cS);
  att[i] /= fmaxf(n1[b * cS + p] * n2[b * cS + q], EPSf);
}

__global__ void attsum_k(const float* __restrict__ att, float* __restrict__ rowsum,
                         float* __restrict__ colsum) {
  int i = blockIdx.x * blockDim.x + threadIdx.x;
  if (i >= cB * cS) return;
  const int b = i / cS, r = i % cS;
  const float* ab = att + (long)b * cS * cS;
  float rs = 0.f, csm = 0.f;
  for (int q = 0; q < cS; ++q) rs += ab[r * cS + q];
  for (int p = 0; p < cS; ++p) csm += ab[p * cS + r];
  rowsum[i] = rs;
  colsum[i] = csm;
}

__global__ void meandiv_k(float* __restrict__ x, const float* __restrict__ sums) {
  int i = blockIdx.x * blockDim.x + threadIdx.x;
  if (i >= cBS * cH) return;
  x[i] /= fmaxf(sums[i / cH], EPSf);
}

// mode 0: out[b,p,h] = max_q v[b,q,h]*att[b,p,q]
// mode 1: out[b,q,h] = max_p v[b,p,h]*att[b,p,q]
__global__ void maxfeat_k(const float* __restrict__ att, const float* __restrict__ v,
                          float* __restrict__ out, int mode) {
  int i = blockIdx.x * blockDim.x + threadIdx.x;
  if (i >= cBS * cH) return;
  const int h = i % cH;
  const int rest = i / cH;
  const int idx = rest % cS;
  const int b = rest / cS;
  const float* ab = att + (long)b * cS * cS;
  const float* vb = v + (long)b * cS * cH;
  float m = -3.402823e38f;
  if (mode == 0)
    for (int q = 0; q < cS; ++q) m = fmaxf(m, vb[q * cH + h] * ab[idx * cS + q]);
  else
    for (int p = 0; p < cS; ++p) m = fmaxf(m, vb[p * cH + h] * ab[p * cS + idx]);
  out[i] = m;
}

// multi-perspective cosine match -> writes L cols at outOff into mv [BS, 6L]
__global__ void mpmatch_k(const float* __restrict__ v1, const float* __restrict__ v2,
                          long v2_bs, long v2_ss, const float* __restrict__ w,
                          float* __restrict__ out, int outLd, int outOff) {
  int i = blockIdx.x * blockDim.x + threadIdx.x;
  if (i >= cBS * cL) return;
  const int l = i % cL;
  const int bs = i / cL;
  const int s = bs % cS, b = bs / cS;
  const float* x1 = v1 + (long)bs * cH;
  const float* x2 = v2 + (long)b * v2_bs + (long)s * v2_ss;
  const float* wl = w + (long)l * cH;
  float num = 0.f, s1 = 0.f, s2 = 0.f;
  for (int h = 0; h < cH; ++h) {
    const float ws = wl[h] * wl[h];
    const float a = x1[h], c = x2[h];
    num += a * c * ws;
    s1 += a * a * ws;
    s2 += c * c * ws;
  }
  out[(long)bs * outLd + outOff + l] = num / fmaxf(sqrtf(s1) * sqrtf(s2), EPSf);
}

// build [hf_p_last | hb_p_first | hf_h_last | hb_h_first] -> [B, 4H]
__global__ void extract_k(const float* __restrict__ hsPf, const float* __restrict__ hsPb,
                          const float* __restrict__ hsHf, const float* __restrict__ hsHb,
                          float* __restrict__ out) {
  int i = blockIdx.x * blockDim.x + threadIdx.x;
  if (i >= cB * c4H) return;
  const int n = i % c4H, b = i / c4H;
  const int which = n / cH, j = n % cH;
  const float* src;
  long idx;
  if (which == 0)      { src = hsPf; idx = ((long)b * cS + (cS - 1)) * cH + j; }
  else if (which == 1) { src = hsPb; idx = ((long)b * cS + 0) * cH + j; }
  else if (which == 2) { src = hsHf; idx = ((long)b * cS + (cS - 1)) * cH + j; }
  else                 { src = hsHb; idx = ((long)b * cS + 0) * cH + j; }
  out[i] = src[idx];
}

__global__ void highway_k(const float* __restrict__ x, const float* __restrict__ lw,
                          const float* __restrict__ lb, const float* __restrict__ gw,
                          const float* __restrict__ gb, float* __restrict__ out,
                          int off) {
  int i = blockIdx.x * blockDim.x + threadIdx.x;
  if (i >= cB * c4H) return;
  const int n = i % c4H, m = i / c4H;
  const float* xr = x + (long)m * c4H;
  const float* lr = lw + (long)n * c4H;
  const float* gr = gw + (long)n * c4H;
  float lin = lb[n], gate = gb[n];
  for (int k = 0; k < c4H; ++k) {
    const float xv = xr[k];
    lin += xv * lr[k];
    gate += xv * gr[k];
  }
  const float hl = lin > 0.f ? lin : 0.f;
  const float t = sigf(gate);
  out[(long)m * (2 * c4H) + off + n] = t * hl + (1.f - t) * xr[n];
}

__global__ void linear_k(const float* __restrict__ x, const float* __restrict__ W,
                         const float* __restrict__ bias, float* __restrict__ out,
                         int M, int N, int K, int act) {
  int i = blockIdx.x * blockDim.x + threadIdx.x;
  if (i >= M * N) return;
  const int n = i % N, m = i / N;
  const float* xr = x + (long)m * K;
  const float* wr = W + (long)n * K;
  float s = bias[n];
  for (int k = 0; k < K; ++k) s += xr[k] * wr[k];
  if (act == 1) s = tanh_fast(s);
  out[i] = s;
}

// ---------------- host-side helpers ----------------
static inline void gemm(hipStream_t st, const float* A, long Ars, long Acs,
                        const float* Bm, long Bks, long Bns, float* C, long ldc,
                        int M, int N, int K, int batch, long sA, long sB, long sC) {
  const int tiles = ((M + 15) / 16) * ((N + 15) / 16);
  dim3 g(batch, (tiles + 7) / 8);
  if (Acs == 1 && Bks == 1)
    wmma_gemm_k<true, true><<<g, 256, 0, st>>>(A, Ars, Acs, Bm, Bks, Bns, C, ldc, M, N, K, sA, sB, sC);
  else if (Acs == 1)
    wmma_gemm_k<true, false><<<g, 256, 0, st>>>(A, Ars, Acs, Bm, Bks, Bns, C, ldc, M, N, K, sA, sB, sC);
  else if (Bks == 1)
    wmma_gemm_k<false, true><<<g, 256, 0, st>>>(A, Ars, Acs, Bm, Bks, Bns, C, ldc, M, N, K, sA, sB, sC);
  else
    wmma_gemm_k<false, false><<<g, 256, 0, st>>>(A, Ars, Acs, Bm, Bks, Bns, C, ldc, M, N, K, sA, sB, sC);
}
#define ELT(n) (((n) + 255) / 256), 256, 0, stream

static void run_level(hipStream_t stream, float* ws,
                      const int* idsP, const int* idsH, const float* emb, int E,
                      const float* WihF, const float* WhhF, const float* bF,
                      const float* WihB, const float* WhhB, const float* bB,
                      const float* aWihF, const float* aWhhF, const float* abF,
                      const float* aWihB, const float* aWhhB, const float* abB,
                      const float* w1, const float* w2, const float* w3,
                      const float* w4, const float* w5, const float* w6,
                      float* outvec) {
  float* embP = ws + O_EMB_P;
  float* embH = ws + O_EMB_H;
  float* xwPf = ws + O_XW;
  float* xwPb = xwPf + SZ_XW;
  float* xwHf = xwPb + SZ_XW;
  float* xwHb = xwHf + SZ_XW;
  float* hsPf = ws + O_HS;
  float* hsPb = hsPf + SZ_HS;
  float* hsHf = hsPb + SZ_HS;
  float* hsHb = hsHf + SZ_HS;
  float* nPf = ws + O_NORM;
  float* nPb = nPf + cBS;
  float* nHf = nPb + cBS;
  float* nHb = nHf + cBS;
  float* attF = ws + O_ATT;
  float* attB = attF + SZ_ATT;
  float* rowF = ws + O_SUM;
  float* colF = rowF + cBS;
  float* rowB = colF + cBS;
  float* colB = rowB + cBS;
  float* meanHf = ws + O_FEAT;
  float* meanPf = meanHf + SZ_HS;
  float* maxHf  = meanPf + SZ_HS;
  float* maxPf  = maxHf + SZ_HS;
  float* meanHb = maxPf + SZ_HS;
  float* meanPb = meanHb + SZ_HS;
  float* maxHb  = meanPb + SZ_HS;
  float* maxPb  = maxHb + SZ_HS;
  float* mvP = ws + O_MV;
  float* mvH = mvP + SZ_MV;

  // pad feature dim to a multiple of 4 so the WMMA K-loop has no tail
  const int Epad = (E + 3) & ~3;
  const float* WF = WihF;
  const float* WB = WihB;
  if (Epad != E) {
    float* wpF = ws + O_WPADF;
    float* wpB = ws + O_WPADB;
    padw_k<<<ELT(c4H * Epad)>>>(WihF, wpF, E, Epad);
    padw_k<<<ELT(c4H * Epad)>>>(WihB, wpB, E, Epad);
    WF = wpF;
    WB = wpB;
  }
  // 1) embedding gathers (zero-padded to Epad)
  gather_k<<<ELT((long)cBS * Epad)>>>(idsP, emb, embP, E, Epad);
  gather_k<<<ELT((long)cBS * Epad)>>>(idsH, emb, embH, E, Epad);
  // 2) input projections x @ Wih^T  (WMMA, NT: both operands k-contiguous)
  gemm(stream, embP, Epad, 1, WF, 1, Epad, xwPf, c4H, cBS, c4H, Epad, 1, 0, 0, 0);
  gemm(stream, embP, Epad, 1, WB, 1, Epad, xwPb, c4H, cBS, c4H, Epad, 1, 0, 0, 0);
  gemm(stream, embH, Epad, 1, WF, 1, Epad, xwHf, c4H, cBS, c4H, Epad, 1, 0, 0, 0);
  gemm(stream, embH, Epad, 1, WB, 1, Epad, xwHb, c4H, cBS, c4H, Epad, 1, 0, 0, 0);
  // 3) context bi-LSTM recurrences (4 parallel blocks, TDM + WMMA per step)
  lstm_rec_k<<<4, 256, 0, stream>>>(xwPf, xwPb, xwHf, xwHb, WhhF, bF, WhhB, bB,
                                    hsPf, hsPb, hsHf, hsHb);
  // 4) norms
  norm_k<<<ELT(cBS)>>>(hsPf, nPf);
  norm_k<<<ELT(cBS)>>>(hsPb, nPb);
  norm_k<<<ELT(cBS)>>>(hsHf, nHf);
  norm_k<<<ELT(cBS)>>>(hsHb, nHb);
  // 5) attention scores (batched WMMA, A @ B^T) then cosine normalize
  gemm(stream, hsPf, cH, 1, hsHf, 1, cH, attF, cS, cS, cS, cH, cB,
       (long)cS * cH, (long)cS * cH, (long)cS * cS);
  gemm(stream, hsPb, cH, 1, hsHb, 1, cH, attB, cS, cS, cS, cH, cB,
       (long)cS * cH, (long)cS * cH, (long)cS * cS);
  attdiv_k<<<ELT((long)cB * cS * cS)>>>(attF, nPf, nHf);
  attdiv_k<<<ELT((long)cB * cS * cS)>>>(attB, nPb, nHb);
  attsum_k<<<ELT(cB * cS)>>>(attF, rowF, colF);
  attsum_k<<<ELT(cB * cS)>>>(attB, rowB, colB);
  // 6) mean features: att @ h (NN) and att^T @ p (strided-A), batched WMMA
  gemm(stream, attF, cS, 1, hsHf, cH, 1, meanHf, cH, cS, cH, cS, cB,
       (long)cS * cS, (long)cS * cH, (long)cS * cH);
  gemm(stream, attF, 1, cS, hsPf, cH, 1, meanPf, cH, cS, cH, cS, cB,
       (long)cS * cS, (long)cS * cH, (long)cS * cH);
  gemm(stream, attB, cS, 1, hsHb, cH, 1, meanHb, cH, cS, cH, cS, cB,
       (long)cS * cS, (long)cS * cH, (long)cS * cH);
  gemm(stream, attB, 1, cS, hsPb, cH, 1, meanPb, cH, cS, cH, cS, cB,
       (long)cS * cS, (long)cS * cH, (long)cS * cH);
  meandiv_k<<<ELT(cBS * cH)>>>(meanHf, rowF);
  meandiv_k<<<ELT(cBS * cH)>>>(meanPf, colF);
  meandiv_k<<<ELT(cBS * cH)>>>(meanHb, rowB);
  meandiv_k<<<ELT(cBS * cH)>>>(meanPb, colB);
  // 7) max features
  maxfeat_k<<<ELT(cBS * cH)>>>(attF, hsHf, maxHf, 0);
  maxfeat_k<<<ELT(cBS * cH)>>>(attF, hsPf, maxPf, 1);
  maxfeat_k<<<ELT(cBS * cH)>>>(attB, hsHb, maxHb, 0);
  maxfeat_k<<<ELT(cBS * cH)>>>(attB, hsPb, maxPb, 1);
  // 8) multi-perspective matching -> mvP / mvH [BS, 120]
  const long SH = (long)cS * cH;
  const int MT = cBS * cL;
  // mv_p
  mpmatch_k<<<ELT(MT)>>>(hsPf, hsHf + (long)(cS - 1) * cH, SH, 0, w1, mvP, 6 * cL, 0);
  mpmatch_k<<<ELT(MT)>>>(hsPf, meanHf, SH, cH, w3, mvP, 6 * cL, cL);
  mpmatch_k<<<ELT(MT)>>>(hsPf, maxHf, SH, cH, w5, mvP, 6 * cL, 2 * cL);
  mpmatch_k<<<ELT(MT)>>>(hsPb, hsHb, SH, 0, w2, mvP, 6 * cL, 3 * cL);
  mpmatch_k<<<ELT(MT)>>>(hsPb, meanHb, SH, cH, w4, mvP, 6 * cL, 4 * cL);
  mpmatch_k<<<ELT(MT)>>>(hsPb, maxHb, SH, cH, w6, mvP, 6 * cL, 5 * cL);
  // mv_h
  mpmatch_k<<<ELT(MT)>>>(hsHf, hsPf + (long)(cS - 1) * cH, SH, 0, w1, mvH, 6 * cL, 0);
  mpmatch_k<<<ELT(MT)>>>(hsHf, meanPf, SH, cH, w3, mvH, 6 * cL, cL);
  mpmatch_k<<<ELT(MT)>>>(hsHf, maxPf, SH, cH, w5, mvH, 6 * cL, 2 * cL);
  mpmatch_k<<<ELT(MT)>>>(hsHb, hsPb, SH, 0, w2, mvH, 6 * cL, 3 * cL);
  mpmatch_k<<<ELT(MT)>>>(hsHb, meanPb, SH, cH, w4, mvH, 6 * cL, 4 * cL);
  mpmatch_k<<<ELT(MT)>>>(hsHb, maxPb, SH, cH, w6, mvH, 6 * cL, 5 * cL);
  // 9) aggregation LSTM: input projections (K=120) reuse xw buffers
  gemm(stream, mvP, 6 * cL, 1, aWihF, 1, 6 * cL, xwPf, c4H, cBS, c4H, 6 * cL, 1, 0, 0, 0);
  gemm(stream, mvP, 6 * cL, 1, aWihB, 1, 6 * cL, xwPb, c4H, cBS, c4H, 6 * cL, 1, 0, 0, 0);
  gemm(stream, mvH, 6 * cL, 1, aWihF, 1, 6 * cL, xwHf, c4H, cBS, c4H, 6 * cL, 1, 0, 0, 0);
  gemm(stream, mvH, 6 * cL, 1, aWihB, 1, 6 * cL, xwHb, c4H, cBS, c4H, 6 * cL, 1, 0, 0, 0);
  lstm_rec_k<<<4, 256, 0, stream>>>(xwPf, xwPb, xwHf, xwHb, aWhhF, abF, aWhhB, abB,
                                    hsPf, hsPb, hsHf, hsHb);
  // 10) last-state extraction -> [B, 4H]
  extract_k<<<ELT(cB * c4H)>>>(hsPf, hsPb, hsHf, hsHb, outvec);
}

extern "C" void kernel_launch(void* const* d_in, const int* in_sizes, int n_in,
                              void* d_out, int out_size, void* d_ws, size_t ws_size,
                              hipStream_t stream) {
  (void)in_sizes; (void)n_in; (void)out_size; (void)ws_size;
  const int* p_ids  = (const int*)d_in[0];
  const int* h_ids  = (const int*)d_in[1];
  const int* cp_ids = (const int*)d_in[2];
  const int* ch_ids = (const int*)d_in[3];
  const float* word_emb = (const float*)d_in[4];
  const float* char_emb = (const float*)d_in[5];
  const float* ctx_Wih_f = (const float*)d_in[6];
  const float* ctx_Whh_f = (const float*)d_in[7];
  const float* ctx_b_f   = (const float*)d_in[8];
  const float* ctx_Wih_b = (const float*)d_in[9];
  const float* ctx_Whh_b = (const float*)d_in[10];
  const float* ctx_b_b   = (const float*)d_in[11];
  const float* chr_Wih_f = (const float*)d_in[12];
  const float* chr_Whh_f = (const float*)d_in[13];
  const float* chr_b_f   = (const float*)d_in[14];
  const float* chr_Wih_b = (const float*)d_in[15];
  const float* chr_Whh_b = (const float*)d_in[16];
  const float* chr_b_b   = (const float*)d_in[17];
  const float* agg_Wih_f = (const float*)d_in[18];
  const float* agg_Whh_f = (const float*)d_in[19];
  const float* agg_b_f   = (const float*)d_in[20];
  const float* agg_Wih_b = (const float*)d_in[21];
  const float* agg_Whh_b = (const float*)d_in[22];
  const float* agg_b_b   = (const float*)d_in[23];
  const float* mp_w1 = (const float*)d_in[24];
  const float* mp_w2 = (const float*)d_in[25];
  const float* mp_w3 = (const float*)d_in[26];
  const float* mp_w4 = (const float*)d_in[27];
  const float* mp_w5 = (const float*)d_in[28];
  const float* mp_w6 = (const float*)d_in[29];
  const float* char_w1 = (const float*)d_in[30];
  const float* char_w2 = (const float*)d_in[31];
  const float* hw_lin_w  = (const float*)d_in[32];
  const float* hw_lin_b  = (const float*)d_in[33];
  const float* hw_gate_w = (const float*)d_in[34];
  const float* hw_gate_b = (const float*)d_in[35];
  const float* fc1_w = (const float*)d_in[36];
  const float* fc1_b = (const float*)d_in[37];
  const float* fc2_w = (const float*)d_in[38];
  const float* fc2_b = (const float*)d_in[39];
  float* ws = (float*)d_ws;
  float* wx = ws + O_WX;
  float* cx = ws + O_CX;
  float* x8 = ws + O_X8;
  float* fc1o = ws + O_FC1;

  // word level -> wx
  run_level(stream, ws, p_ids, h_ids, word_emb, cE,
            ctx_Wih_f, ctx_Whh_f, ctx_b_f, ctx_Wih_b, ctx_Whh_b, ctx_b_b,
            agg_Wih_f, agg_Whh_f, agg_b_f, agg_Wih_b, agg_Whh_b, agg_b_b,
            mp_w1, mp_w2, mp_w3, mp_w4, mp_w5, mp_w6, wx);
  // char level -> cx  (reuses all large scratch)
  run_level(stream, ws, cp_ids, ch_ids, char_emb, cCD,
            chr_Wih_f, chr_Whh_f, chr_b_f, chr_Wih_b, chr_Whh_b, chr_b_b,
            agg_Wih_f, agg_Whh_f, agg_b_f, agg_Wih_b, agg_Whh_b, agg_b_b,
            char_w1, char_w2, mp_w3, mp_w4, mp_w5, mp_w6, cx);
  // head: highway -> concat -> fc1(tanh) -> fc2
  highway_k<<<ELT(cB * c4H)>>>(wx, hw_lin_w, hw_lin_b, hw_gate_w, hw_gate_b, x8, 0);
  highway_k<<<ELT(cB * c4H)>>>(cx, hw_lin_w, hw_lin_b, hw_gate_w, hw_gate_b, x8, c4H);
  linear_k<<<ELT(cB * 2 * cH)>>>(x8, fc1_w, fc1_b, fc1o, cB, 2 * cH, 2 * c4H, 1);
  linear_k<<<ELT(cB * cCLS)>>>(fc1o, fc2_w, fc2_b, (float*)d_out, cB, cCLS, 2 * cH, 0);
}